// Net_68994354643186
// MI455X (gfx1250) — compile-verified
//
#include <hip/hip_runtime.h>
#include <hip/hip_bf16.h>
#include <math.h>

typedef __attribute__((ext_vector_type(2))) float v2f;
typedef __attribute__((ext_vector_type(8))) float v8f;

#define NHEADS 4
#define NEG_INF_KEY 0x007FFFFFu  // ordered-key encoding of -inf

// ---- monotonic float<->uint key (atomicMax over uint == max over float) ----
__device__ __forceinline__ unsigned f2key(float f) {
    unsigned b = __float_as_uint(f);
    return (b & 0x80000000u) ? ~b : (b | 0x80000000u);
}
__device__ __forceinline__ float key2f(unsigned k) {
    unsigned b = (k & 0x80000000u) ? (k ^ 0x80000000u) : ~k;
    return __uint_as_float(b);
}

// ---------------------------------------------------------------------------
// Pack W[K,N] (+ zero padding) into WMMA-B-fragment order:
//   Wp[((kq*Npad + col) << 1) + j] = { W[4kq+2j][col], W[4kq+2j+1][col] }
// so a lane's B fragment is one unguarded 8-byte load.
// ---------------------------------------------------------------------------
__global__ void pack_w(const float* __restrict__ W, v2f* __restrict__ Wp,
                       int K, int N, int Kq, int Npad)
{
    int t = blockIdx.x * blockDim.x + threadIdx.x;
    if (t >= Kq * Npad) return;
    int kq = t / Npad;
    int c  = t - kq * Npad;
    float vals[4];
#pragma unroll
    for (int jj = 0; jj < 4; ++jj) {
        int kk = 4 * kq + jj;
        vals[jj] = (c < N && kk < K) ? W[(size_t)kk * N + c] : 0.0f;
    }
    v2f lo; lo.x = vals[0]; lo.y = vals[1];
    v2f hi; hi.x = vals[2]; hi.y = vals[3];
    Wp[((size_t)t << 1) + 0] = lo;
    Wp[((size_t)t << 1) + 1] = hi;
}

// ---------------------------------------------------------------------------
// Y[M,N] = X[M,K] @ W[K,N] + bias, W pre-packed/padded (Npad % 32 == 0).
// One wave -> 16x32 output (two 16x16 f32 tiles) via V_WMMA_F32_16X16X4_F32.
// M % 16 == 0 (50000 = 3125*16).  Inner loop is branch-free: one b64 A load,
// two b64 B loads, two WMMAs per K-step.  K-tail (K%4 != 0) handled once.
// A frag (16x4): lane l holds row l%16, K = 2*(l/16)+{0,1}.
// B frag (4x16): lane l holds col l%16, K = 2*(l/16)+{0,1}.
// C/D: vgpr r, lane l -> (M = r + 8*(l/16), N = l%16).
// ---------------------------------------------------------------------------
__global__ void gemm_bias_wmma(const float* __restrict__ X,
                               const v2f* __restrict__ Wp,
                               const float* __restrict__ bias,
                               float* __restrict__ Y,
                               int M, int K, int N, int Npad)
{
    const int lane = threadIdx.x & 31;
    const int wave = threadIdx.x >> 5;
    const int NT2  = Npad >> 5;               // 32-wide column groups
    const int tile = blockIdx.x * 8 + wave;   // 256 threads = 8 waves/block
    const int Mt   = M >> 4;
    if (tile >= Mt * NT2) return;             // wave-uniform: EXEC stays full

    const int mt   = tile / NT2;
    const int ntp  = tile - mt * NT2;
    const int l15  = lane & 15;
    const int arow = (mt << 4) + l15;
    const int col0 = (ntp << 5) + l15;
    const int col1 = col0 + 16;
    const int khalf = (lane >> 4) << 1;       // 0 or 2
    const int jsel  = lane >> 4;              // 0 or 1

    const float* __restrict__ xrow = X + (size_t)arow * K;
    v8f acc0 = {}, acc1 = {};
    const int K4 = K & ~3;
    for (int k0 = 0; k0 < K4; k0 += 4) {
        const int kq = k0 >> 2;
        v2f a  = *(const v2f*)(xrow + k0 + khalf);
        v2f b0 = Wp[(((size_t)kq * Npad + col0) << 1) + jsel];
        v2f b1 = Wp[(((size_t)kq * Npad + col1) << 1) + jsel];
        acc0 = __builtin_amdgcn_wmma_f32_16x16x4_f32(
            false, a, false, b0, (short)0, acc0, false, false);
        acc1 = __builtin_amdgcn_wmma_f32_16x16x4_f32(
            false, a, false, b1, (short)0, acc1, false, false);
    }
    if (K4 < K) {                             // once per tile (layer 1, K=130)
        const int kq  = K4 >> 2;
        const int ka0 = K4 + khalf, ka1 = ka0 + 1;
        v2f a;
        a.x = (ka0 < K) ? xrow[ka0] : 0.0f;
        a.y = (ka1 < K) ? xrow[ka1] : 0.0f;
        v2f b0 = Wp[(((size_t)kq * Npad + col0) << 1) + jsel];
        v2f b1 = Wp[(((size_t)kq * Npad + col1) << 1) + jsel];
        acc0 = __builtin_amdgcn_wmma_f32_16x16x4_f32(
            false, a, false, b0, (short)0, acc0, false, false);
        acc1 = __builtin_amdgcn_wmma_f32_16x16x4_f32(
            false, a, false, b1, (short)0, acc1, false, false);
    }

    const int crow = (mt << 4) + ((lane >> 4) << 3);
    if (col0 < N) {
        const float bv = bias[col0];
#pragma unroll
        for (int r = 0; r < 8; ++r)
            Y[(size_t)(crow + r) * N + col0] = acc0[r] + bv;
    }
    if (col1 < N) {
        const float bv = bias[col1];
#pragma unroll
        for (int r = 0; r < 8; ++r)
            Y[(size_t)(crow + r) * N + col1] = acc1[r] + bv;
    }
}

// ---------------------------------------------------------------------------
__global__ void fill_u32(unsigned* __restrict__ p, unsigned val, int n)
{
    int i = blockIdx.x * blockDim.x + threadIdx.x;
    if (i < n) p[i] = val;
}

// Pass A: one thread per edge.  float4 gathers of whole q[dst]/k[src] rows
// (16B-aligned, L2-resident); 4 head dots; segment-max via ordered-uint
// atomicMax.  (4i+j)/D constant-folds under full unroll.
template<int D>
__global__ void edge_scores_kernel(const float* __restrict__ q,
                                   const float* __restrict__ k,
                                   const int* __restrict__ src,
                                   const int* __restrict__ dst,
                                   float* __restrict__ scores,
                                   unsigned* __restrict__ mkey,
                                   int E, float scale)
{
    int e = blockIdx.x * blockDim.x + threadIdx.x;
    if (e >= E) return;
    const int s = src[e];
    const int d = dst[e];
    const float4* __restrict__ qp = (const float4*)(q + (size_t)d * (NHEADS * D));
    const float4* __restrict__ kp = (const float4*)(k + (size_t)s * (NHEADS * D));
    float acc[NHEADS] = {0.0f, 0.0f, 0.0f, 0.0f};
#pragma unroll
    for (int i = 0; i < D; ++i) {             // HD = 4D floats = D float4 chunks
        float4 qv = qp[i];
        float4 kv = kp[i];
        acc[(4 * i + 0) / D] = fmaf(qv.x, kv.x, acc[(4 * i + 0) / D]);
        acc[(4 * i + 1) / D] = fmaf(qv.y, kv.y, acc[(4 * i + 1) / D]);
        acc[(4 * i + 2) / D] = fmaf(qv.z, kv.z, acc[(4 * i + 2) / D]);
        acc[(4 * i + 3) / D] = fmaf(qv.w, kv.w, acc[(4 * i + 3) / D]);
    }
    float4 sc;
    sc.x = acc[0] * scale; sc.y = acc[1] * scale;
    sc.z = acc[2] * scale; sc.w = acc[3] * scale;
    ((float4*)scores)[e] = sc;
    unsigned* mrow = mkey + (size_t)d * NHEADS;
    atomicMax(mrow + 0, f2key(sc.x));
    atomicMax(mrow + 1, f2key(sc.y));
    atomicMax(mrow + 2, f2key(sc.z));
    atomicMax(mrow + 3, f2key(sc.w));
}

// Pass B: w = exp(score - m[dst]); denom[dst] += w; acc[dst] += w * v[src]
// (unnormalized; one divide per node in finalize).
template<int D>
__global__ void edge_accum_kernel(const float* __restrict__ v,
                                  const float* __restrict__ scores,
                                  const unsigned* __restrict__ mkey,
                                  const int* __restrict__ src,
                                  const int* __restrict__ dst,
                                  float* __restrict__ acc,
                                  float* __restrict__ denom,
                                  int E)
{
    int e = blockIdx.x * blockDim.x + threadIdx.x;
    if (e >= E) return;
    const int s = src[e];
    const int d = dst[e];
    float4 sc = ((const float4*)scores)[e];
    uint4  mk = ((const uint4*)mkey)[d];
    float w[NHEADS];
    w[0] = __expf(sc.x - key2f(mk.x));
    w[1] = __expf(sc.y - key2f(mk.y));
    w[2] = __expf(sc.z - key2f(mk.z));
    w[3] = __expf(sc.w - key2f(mk.w));
    float* drow = denom + (size_t)d * NHEADS;
    atomicAdd(drow + 0, w[0]);
    atomicAdd(drow + 1, w[1]);
    atomicAdd(drow + 2, w[2]);
    atomicAdd(drow + 3, w[3]);
    const float4* __restrict__ vp = (const float4*)(v + (size_t)s * (NHEADS * D));
    float* __restrict__ ap = acc + (size_t)d * (NHEADS * D);
#pragma unroll
    for (int i = 0; i < D; ++i) {
        float4 vv = vp[i];
        atomicAdd(ap + 4 * i + 0, w[(4 * i + 0) / D] * vv.x);
        atomicAdd(ap + 4 * i + 1, w[(4 * i + 1) / D] * vv.y);
        atomicAdd(ap + 4 * i + 2, w[(4 * i + 2) / D] * vv.z);
        atomicAdd(ap + 4 * i + 3, w[(4 * i + 3) / D] * vv.w);
    }
}

// hbuf already holds the skip projection (x @ Ws + bs); add normalized
// attention output, then optional LeakyReLU.
__global__ void finalize_kernel(float* __restrict__ hbuf,
                                const float* __restrict__ acc,
                                const float* __restrict__ denom,
                                int N, int D, int leaky)
{
    int idx = blockIdx.x * blockDim.x + threadIdx.x;
    const int HD = NHEADS * D;
    if (idx >= N * HD) return;
    const int n = idx / HD;
    const int c = idx - n * HD;
    const int hh = c / D;
    float val = hbuf[idx] + acc[idx] / (denom[(size_t)n * NHEADS + hh] + 1e-16f);
    if (leaky) val = (val >= 0.0f) ? val : 0.1f * val;
    hbuf[idx] = val;
}

__global__ void log_softmax_kernel(const float* __restrict__ in,
                                   float* __restrict__ out, int N, int C)
{
    int n = blockIdx.x * blockDim.x + threadIdx.x;
    if (n >= N) return;
    const float* __restrict__ row = in + (size_t)n * C;
    float mx = -INFINITY;
    for (int c = 0; c < C; ++c) mx = fmaxf(mx, row[c]);
    float s = 0.0f;
    for (int c = 0; c < C; ++c) s += __expf(row[c] - mx);
    const float lse = mx + __logf(s);
    float* __restrict__ orow = out + (size_t)n * C;
    for (int c = 0; c < C; ++c) orow[c] = row[c] - lse;
}

// ---------------------------------------------------------------------------
struct LayerBufs {
    float* qb; float* kb; float* vb; float* accb;
    float* scb; unsigned* mkey; float* denom;
    v2f* wp[4];
    const int* src; const int* dst;
    int N; int E;
};

template<int D>
static void run_layer(const float* X, int Cin,
                      const float* const* P, int pbase,  // P[pbase..pbase+7]
                      float* hout, int leaky,
                      const LayerBufs& B, hipStream_t stream)
{
    const int N    = B.N;
    const int E    = B.E;
    const int HD   = NHEADS * D;
    const int Npad = (HD + 31) & ~31;
    const int Kq   = (Cin + 3) >> 2;

    // pack all four weight matrices, then run the four projections
    const int pkThreads = Kq * Npad;
    const int pkBlocks  = (pkThreads + 255) / 256;
    for (int m = 0; m < 4; ++m)
        pack_w<<<pkBlocks, 256, 0, stream>>>(P[pbase + 2 * m], B.wp[m],
                                             Cin, HD, Kq, Npad);

    const int tiles = (N >> 4) * (Npad >> 5);
    const int gblk  = (tiles + 7) / 8;        // 8 waves per 256-thread block
    gemm_bias_wmma<<<gblk, 256, 0, stream>>>(X, B.wp[0], P[pbase + 1], B.qb, N, Cin, HD, Npad);
    gemm_bias_wmma<<<gblk, 256, 0, stream>>>(X, B.wp[1], P[pbase + 3], B.kb, N, Cin, HD, Npad);
    gemm_bias_wmma<<<gblk, 256, 0, stream>>>(X, B.wp[2], P[pbase + 5], B.vb, N, Cin, HD, Npad);
    gemm_bias_wmma<<<gblk, 256, 0, stream>>>(X, B.wp[3], P[pbase + 7], hout, N, Cin, HD, Npad);

    fill_u32<<<(N * NHEADS + 255) / 256, 256, 0, stream>>>(B.mkey, NEG_INF_KEY, N * NHEADS);
    fill_u32<<<(N * NHEADS + 255) / 256, 256, 0, stream>>>((unsigned*)B.denom, 0u, N * NHEADS);
    fill_u32<<<(N * HD + 255) / 256, 256, 0, stream>>>((unsigned*)B.accb, 0u, N * HD);

    const float scale = 1.0f / sqrtf((float)D);
    edge_scores_kernel<D><<<(E + 255) / 256, 256, 0, stream>>>(
        B.qb, B.kb, B.src, B.dst, B.scb, B.mkey, E, scale);
    edge_accum_kernel<D><<<(E + 255) / 256, 256, 0, stream>>>(
        B.vb, B.scb, B.mkey, B.src, B.dst, B.accb, B.denom, E);
    finalize_kernel<<<(N * HD + 255) / 256, 256, 0, stream>>>(
        hout, B.accb, B.denom, N, D, leaky);
}

// ---------------------------------------------------------------------------
extern "C" void kernel_launch(void* const* d_in, const int* in_sizes, int n_in,
                              void* d_out, int out_size, void* d_ws, size_t ws_size,
                              hipStream_t stream)
{
    (void)n_in; (void)out_size; (void)ws_size;
    const int N = 50000;
    const int E = in_sizes[1] / 2;

    const float* x   = (const float*)d_in[0];
    const int*   ei  = (const int*)d_in[1];

    // ---- workspace layout (256B aligned slices) ----
    char* ws = (char*)d_ws;
    size_t off = 0;
    auto carve = [&](size_t bytes) -> char* {
        char* p = ws + off;
        off += (bytes + 255) & ~(size_t)255;
        return p;
    };
    LayerBufs B;
    B.qb    = (float*)carve((size_t)N * 200 * sizeof(float));
    B.kb    = (float*)carve((size_t)N * 200 * sizeof(float));
    B.vb    = (float*)carve((size_t)N * 200 * sizeof(float));
    B.accb  = (float*)carve((size_t)N * 200 * sizeof(float));
    B.scb   = (float*)carve((size_t)E * NHEADS * sizeof(float));
    B.mkey  = (unsigned*)carve((size_t)N * NHEADS * sizeof(unsigned));
    B.denom = (float*)carve((size_t)N * NHEADS * sizeof(float));
    for (int m = 0; m < 4; ++m)
        B.wp[m] = (v2f*)carve((size_t)50 * 224 * 2 * sizeof(v2f));  // max Kq*Npad*2
    float* h1 = (float*)carve((size_t)N * 200 * sizeof(float));
    float* h2 = (float*)carve((size_t)N * 100 * sizeof(float));
    float* h3 = h1;  // h1 is dead once layer-2 projections complete
    B.src = ei;
    B.dst = ei + E;
    B.N = N;
    B.E = E;

    const float* const* P = (const float* const*)d_in;
    // layer 1: 130 -> 4x50 (=200), LeakyReLU
    run_layer<50>(x,  130, P, 2,  h1, 1, B, stream);
    // layer 2: 200 -> 4x25 (=100), LeakyReLU
    run_layer<25>(h1, 200, P, 10, h2, 1, B, stream);
    // layer 3: 100 -> 4x10 (=40), no activation
    run_layer<10>(h2, 100, P, 18, h3, 0, B, stream);

    log_softmax_kernel<<<(N + 255) / 256, 256, 0, stream>>>(h3, (float*)d_out, N, 40);
}